// FastAttention_28879360099122
// MI455X (gfx1250) — compile-verified
//
#include <hip/hip_runtime.h>
#include <hip/hip_bf16.h>

// ---------------- problem constants ----------------
#define DIMH   64
#define NBF    256
#define BATCH  8
#define NHEADS 12
#define SEQ    4096
#define NHT    (BATCH*NHEADS)        // 96 heads total
#define EPSV   1e-4f
#define DN     0.3535533905932738f   // 64^-0.25
#define RATIO  0.0625f               // 256^-0.5

// LDS pitches (floats). 68 % 64 == 4 -> conflict-free ds_load_b64 A/B fetches.
#define KLD    68                    // k_s/q_s/proj_s/v_s/ctx_s pitch
#define KPLD   257                   // k' staging pitch
#define QPLD   260                   // q' staging pitch (260 % 64 == 4, 16B-aligned rows)

// ---------------- workspace layout (bytes) ----------------
#define WS_HEADMAX_OFF 0                           // 96 * u32
#define WS_KSUM_OFF    1024                        // 96*256 f32
#define WS_CTX_OFF     (1024 + NHT*NBF*4)          // 96*256*64 f32
#define WS_TOTAL_BYTES (WS_CTX_OFF + NHT*NBF*DIMH*4)
#define WS_TOTAL_WORDS (WS_TOTAL_BYTES/4)

typedef float v2f __attribute__((ext_vector_type(2)));
typedef float v8f __attribute__((ext_vector_type(8)));

__device__ __forceinline__ v8f wmma4(v2f a, v2f b, v8f c) {
  // D(16x16,f32) = A(16x4,f32) x B(4x16,f32) + C
  return __builtin_amdgcn_wmma_f32_16x16x4_f32(false, a, false, b, (short)0, c, false, false);
}

// order-preserving float <-> uint encoding for atomicMax on floats
__device__ __forceinline__ unsigned fenc(float f) {
  unsigned u = __float_as_uint(f);
  return (u & 0x80000000u) ? ~u : (u | 0x80000000u);
}
__device__ __forceinline__ float fdec(unsigned e) {
  unsigned u = (e & 0x80000000u) ? (e & 0x7FFFFFFFu) : ~e;
  return __uint_as_float(u);
}

// ---------------- kernel 0: zero workspace ----------------
__global__ void fa_init(unsigned* ws) {
  int i = blockIdx.x * blockDim.x + threadIdx.x;
  if (i < WS_TOTAL_WORDS) ws[i] = 0u;   // 0 < fenc(x) for every real x -> valid max identity
}

// ---------------- shared helpers ----------------
__device__ __forceinline__ void load_proj(const float* __restrict__ P, float* proj_s, int tid) {
  const float4* p4 = (const float4*)P;               // proj: [256][64] row-major
  #pragma unroll
  for (int i = 0; i < 16; ++i)
    *(float4*)&proj_s[tid * KLD + i * 4] = p4[tid * 16 + i];
}

// ---------------- kernel 1: per-head max of k_dash ----------------
__global__ __launch_bounds__(256) void fa_kmax(const float* __restrict__ K,
                                               const float* __restrict__ P,
                                               unsigned* __restrict__ headmax) {
  __shared__ float proj_s[NBF * KLD];
  __shared__ float k_s[64 * KLD];
  __shared__ float wmax_s[8];
  const int tid = threadIdx.x, lane = tid & 31, w = tid >> 5;
  const int half = lane >> 4, lm = lane & 15;
  const int head = blockIdx.y;
  const int n0 = blockIdx.x * 64;

  load_proj(P, proj_s, tid);
  {
    const int r = tid >> 2, q4 = tid & 3;
    const float4* kg = (const float4*)(K + (size_t)head * SEQ * DIMH + (size_t)n0 * DIMH);
    #pragma unroll
    for (int j = 0; j < 4; ++j) {
      float4 kv = kg[r * 16 + q4 * 4 + j];
      kv.x *= DN; kv.y *= DN; kv.z *= DN; kv.w *= DN;
      *(float4*)&k_s[r * KLD + q4 * 16 + j * 4] = kv;
    }
  }
  __syncthreads();

  float m = -3.4e38f;
  for (int i = 0; i < 8; ++i) {                      // uniform trip count: EXEC stays full
    const int t = w + 8 * i;
    const int r = t & 3, c = t >> 2;
    v8f acc = {};
    const float* ar = &k_s[(r * 16 + lm) * KLD];
    const float* br = &proj_s[(c * 16 + lm) * KLD];
    #pragma unroll
    for (int k0 = 0; k0 < 64; k0 += 4) {
      v2f a = *(const v2f*)(ar + k0 + half * 2);
      v2f b = *(const v2f*)(br + k0 + half * 2);
      acc = wmma4(a, b, acc);
    }
    #pragma unroll
    for (int vi = 0; vi < 8; ++vi) m = fmaxf(m, acc[vi]);
  }
  #pragma unroll
  for (int off = 16; off; off >>= 1) m = fmaxf(m, __shfl_xor(m, off, 32));
  if (lane == 0) wmax_s[w] = m;
  __syncthreads();
  if (tid == 0) {
    float bm = wmax_s[0];
    #pragma unroll
    for (int i = 1; i < 8; ++i) bm = fmaxf(bm, wmax_s[i]);
    atomicMax(&headmax[head], fenc(bm));
  }
}

// ---------------- kernel 2: k' features, context += k'^T V, k_cumsum ----------------
__global__ __launch_bounds__(256) void fa_ctx(const float* __restrict__ K,
                                              const float* __restrict__ V,
                                              const float* __restrict__ P,
                                              const unsigned* __restrict__ headmax,
                                              float* __restrict__ ksum_g,
                                              float* __restrict__ ctx_g) {
  __shared__ float proj_s[NBF * KLD];
  __shared__ float k_s[64 * KLD];
  __shared__ float v_s[64 * KLD];
  __shared__ float kp_s[64 * KPLD];
  __shared__ float ssq_s[64];
  const int tid = threadIdx.x, lane = tid & 31, w = tid >> 5;
  const int half = lane >> 4, lm = lane & 15;
  const int head = blockIdx.y, chunk = blockIdx.x;   // 8 chunks of 512 rows
  const float stab = fdec(headmax[head]);
  const float* Kh = K + (size_t)head * SEQ * DIMH;
  const float* Vh = V + (size_t)head * SEQ * DIMH;

  load_proj(P, proj_s, tid);

  v8f cacc[8];
  #pragma unroll
  for (int i = 0; i < 8; ++i) cacc[i] = (v8f){};
  float ksum_acc = 0.f;

  for (int sb = 0; sb < 8; ++sb) {
    const int n0 = chunk * 512 + sb * 64;
    __syncthreads();                                  // protect LDS reuse
    {
      const int r = tid >> 2, q4 = tid & 3;
      const float4* kg = (const float4*)(Kh + (size_t)n0 * DIMH);
      const float4* vg = (const float4*)(Vh + (size_t)n0 * DIMH);
      float ss = 0.f;
      #pragma unroll
      for (int j = 0; j < 4; ++j) {
        float4 kv = kg[r * 16 + q4 * 4 + j];
        kv.x *= DN; kv.y *= DN; kv.z *= DN; kv.w *= DN;
        ss += kv.x * kv.x + kv.y * kv.y + kv.z * kv.z + kv.w * kv.w;
        *(float4*)&k_s[r * KLD + q4 * 16 + j * 4] = kv;
        *(float4*)&v_s[r * KLD + q4 * 16 + j * 4] = vg[r * 16 + q4 * 4 + j];
      }
      ss += __shfl_xor(ss, 1, 32); ss += __shfl_xor(ss, 2, 32);
      if (q4 == 0) ssq_s[r] = ss;                    // = dn^2 * ||k_row||^2
    }
    // prefetch next sub-block's k/v (global_prefetch_b8): one cacheline per thread
    if (sb + 1 < 8) {
      const char* nk = (const char*)(Kh + (size_t)(n0 + 64) * DIMH);
      const char* nv = (const char*)(Vh + (size_t)(n0 + 64) * DIMH);
      __builtin_prefetch(nk + tid * 64, 0, 3);
      __builtin_prefetch(nv + tid * 64, 0, 3);
    }
    __syncthreads();

    // k_dash tiles -> k' in LDS (uniform 8-trip loop)
    for (int i = 0; i < 8; ++i) {
      const int t = w + 8 * i;
      const int r = t & 3, c = t >> 2;
      v8f acc = {};
      const float* ar = &k_s[(r * 16 + lm) * KLD];
      const float* br = &proj_s[(c * 16 + lm) * KLD];
      #pragma unroll
      for (int k0 = 0; k0 < 64; k0 += 4) {
        v2f a = *(const v2f*)(ar + k0 + half * 2);
        v2f b = *(const v2f*)(br + k0 + half * 2);
        acc = wmma4(a, b, acc);
      }
      #pragma unroll
      for (int vi = 0; vi < 8; ++vi) {
        const int row = r * 16 + half * 8 + vi;      // C/D layout: M = vgpr + (lane/16)*8
        const float kp = RATIO * (__expf(acc[vi] - 0.5f * ssq_s[row] - stab) + EPSV);
        kp_s[row * KPLD + c * 16 + lm] = kp;
      }
    }
    __syncthreads();

    // k_cumsum partial (column sums)
    {
      float s = 0.f;
      #pragma unroll 8
      for (int r = 0; r < 64; ++r) s += kp_s[r * KPLD + tid];
      ksum_acc += s;
    }
    // context += k'^T x V  (persistent accumulators; 64 tiles of 16x16 across 8 waves)
    #pragma unroll
    for (int i = 0; i < 8; ++i) {
      const int t = w + 8 * i, mt = t & 15, et = t >> 4;
      v8f acc = cacc[i];
      const int mcol = mt * 16 + lm;
      #pragma unroll
      for (int nn = 0; nn < 64; nn += 4) {
        const int r0 = nn + half * 2;
        v2f a, b;
        a.x = kp_s[r0 * KPLD + mcol];
        a.y = kp_s[(r0 + 1) * KPLD + mcol];
        b.x = v_s[r0 * KLD + et * 16 + lm];
        b.y = v_s[(r0 + 1) * KLD + et * 16 + lm];
        acc = wmma4(a, b, acc);
      }
      cacc[i] = acc;
    }
  }

  atomicAdd(&ksum_g[head * NBF + tid], ksum_acc);
  float* cg = ctx_g + (size_t)head * NBF * DIMH;
  #pragma unroll
  for (int i = 0; i < 8; ++i) {
    const int t = w + 8 * i, mt = t & 15, et = t >> 4;
    #pragma unroll
    for (int vi = 0; vi < 8; ++vi) {
      const int m = mt * 16 + half * 8 + vi;
      const int e = et * 16 + lm;
      atomicAdd(&cg[m * DIMH + e], cacc[i][vi]);
    }
  }
}

// ---------------- kernel 3: q' features, out = (q' ctx) * d_inv ----------------
__global__ __launch_bounds__(256) void fa_out(const float* __restrict__ Q,
                                              const float* __restrict__ P,
                                              const float* __restrict__ ksum_g,
                                              const float* __restrict__ ctx_g,
                                              float* __restrict__ out) {
  __shared__ float proj_s[NBF * KLD];
  __shared__ float q_s[64 * KLD];
  __shared__ float ctx_s[NBF * KLD];
  __shared__ float qp_s[64 * QPLD];
  __shared__ float ssq_s[64];
  __shared__ float ksum_s[NBF];
  __shared__ float dinv_s[64];
  const int tid = threadIdx.x, lane = tid & 31, w = tid >> 5;
  const int half = lane >> 4, lm = lane & 15;
  const int head = blockIdx.y;
  const int n0 = blockIdx.x * 64;

  load_proj(P, proj_s, tid);
  {
    const float4* cg = (const float4*)(ctx_g + (size_t)head * NBF * DIMH);
    #pragma unroll
    for (int i = 0; i < 16; ++i)
      *(float4*)&ctx_s[tid * KLD + i * 4] = cg[tid * 16 + i];
    ksum_s[tid] = ksum_g[head * NBF + tid];
  }
  {
    const int r = tid >> 2, q4 = tid & 3;
    const float4* qg = (const float4*)(Q + (size_t)head * SEQ * DIMH + (size_t)n0 * DIMH);
    float ss = 0.f;
    #pragma unroll
    for (int j = 0; j < 4; ++j) {
      float4 qv = qg[r * 16 + q4 * 4 + j];
      qv.x *= DN; qv.y *= DN; qv.z *= DN; qv.w *= DN;
      ss += qv.x * qv.x + qv.y * qv.y + qv.z * qv.z + qv.w * qv.w;
      *(float4*)&q_s[r * KLD + q4 * 16 + j * 4] = qv;
    }
    ss += __shfl_xor(ss, 1, 32); ss += __shfl_xor(ss, 2, 32);
    if (q4 == 0) ssq_s[r] = ss;
  }
  __syncthreads();

  // q_dash tiles -> raw dash in qp_s (uniform 8-trip loop)
  for (int i = 0; i < 8; ++i) {
    const int t = w + 8 * i;
    const int r = t & 3, c = t >> 2;
    v8f acc = {};
    const float* ar = &q_s[(r * 16 + lm) * KLD];
    const float* br = &proj_s[(c * 16 + lm) * KLD];
    #pragma unroll
    for (int k0 = 0; k0 < 64; k0 += 4) {
      v2f a = *(const v2f*)(ar + k0 + half * 2);
      v2f b = *(const v2f*)(br + k0 + half * 2);
      acc = wmma4(a, b, acc);
    }
    #pragma unroll
    for (int vi = 0; vi < 8; ++vi)
      qp_s[(r * 16 + half * 8 + vi) * QPLD + c * 16 + lm] = acc[vi];
  }
  __syncthreads();

  // row max, exp transform, denominator
  {
    const int row = tid >> 2, q4 = tid & 3, mb = q4 * 64;
    float mx = -3.4e38f;
    #pragma unroll 8
    for (int m = 0; m < 64; ++m) mx = fmaxf(mx, qp_s[row * QPLD + mb + m]);
    mx = fmaxf(mx, __shfl_xor(mx, 1, 32));
    mx = fmaxf(mx, __shfl_xor(mx, 2, 32));
    const float diag = 0.5f * ssq_s[row];
    float dp = 0.f;
    #pragma unroll 8
    for (int m = 0; m < 64; ++m) {
      const float d = qp_s[row * QPLD + mb + m];
      const float val = RATIO * (__expf(d - diag - mx) + EPSV);
      qp_s[row * QPLD + mb + m] = val;
      dp += val * ksum_s[mb + m];
    }
    dp += __shfl_xor(dp, 1, 32); dp += __shfl_xor(dp, 2, 32);
    if (q4 == 0) dinv_s[row] = 1.f / dp;
  }
  __syncthreads();

  // out = q' x ctx, K = 256; 16 tiles -> 2 per wave
  float* og = out + (size_t)head * SEQ * DIMH + (size_t)n0 * DIMH;
  #pragma unroll
  for (int i = 0; i < 2; ++i) {
    const int t = w + 8 * i, rt = t & 3, et = t >> 2;
    v8f acc = {};
    const float* ar = &qp_s[(rt * 16 + lm) * QPLD];
    #pragma unroll 8
    for (int k0 = 0; k0 < 256; k0 += 4) {
      const int kr = k0 + half * 2;
      v2f a = *(const v2f*)(ar + kr);
      v2f b;
      b.x = ctx_s[kr * KLD + et * 16 + lm];
      b.y = ctx_s[(kr + 1) * KLD + et * 16 + lm];
      acc = wmma4(a, b, acc);
    }
    #pragma unroll
    for (int vi = 0; vi < 8; ++vi) {
      const int row = rt * 16 + half * 8 + vi;
      og[(size_t)row * DIMH + et * 16 + lm] = acc[vi] * dinv_s[row];
    }
  }
}

// ---------------- launch ----------------
extern "C" void kernel_launch(void* const* d_in, const int* in_sizes, int n_in,
                              void* d_out, int out_size, void* d_ws, size_t ws_size,
                              hipStream_t stream) {
  const float* q = (const float*)d_in[0];
  const float* k = (const float*)d_in[1];
  const float* v = (const float*)d_in[2];
  const float* p = (const float*)d_in[3];
  float* out = (float*)d_out;

  unsigned* headmax = (unsigned*)((char*)d_ws + WS_HEADMAX_OFF);
  float* ksum = (float*)((char*)d_ws + WS_KSUM_OFF);
  float* ctx = (float*)((char*)d_ws + WS_CTX_OFF);

  fa_init<<<(WS_TOTAL_WORDS + 255) / 256, 256, 0, stream>>>((unsigned*)d_ws);
  dim3 gRows(SEQ / 64, NHT);
  fa_kmax<<<gRows, 256, 0, stream>>>(k, p, headmax);
  dim3 gCtx(8, NHT);
  fa_ctx<<<gCtx, 256, 0, stream>>>(k, v, p, headmax, ksum, ctx);
  fa_out<<<gRows, 256, 0, stream>>>(q, p, ksum, ctx, out);
}